// HardMiningLoss_88021059764796
// MI455X (gfx1250) — compile-verified
//
#include <hip/hip_runtime.h>
#include <hip/hip_bf16.h>

#define NROW 8192
#define DIM  128
#define MARGIN 0.1f
#define NTILE (NROW / 16)   // 512 16x16 tiles per dimension

typedef __attribute__((ext_vector_type(2))) float v2f;
typedef __attribute__((ext_vector_type(8))) float v8f;

// Compute one 16x16 fp32 sim tile (rows = afrag's row tile, cols = colBase..colBase+15)
// using V_WMMA_F32_16X16X4_F32, K-loop over DIM in steps of 4.
// Layout (ISA 7.12.2, 32-bit 16x4 A / 4x16 B, wave32):
//   lane L: m/n = L&15 ; VGPR v holds K = k0 + 2*(L>>4) + v
// B fragment is bulk-staged into registers first so the 32 global_load_b64 issue
// as one clause (single loadcnt wait) instead of load->wait0->wmma serialization.
// Two interleaved accumulators break the WMMA RAW chain.
__device__ __forceinline__ v8f sim_tile(const float* __restrict__ X,
                                        const v2f* __restrict__ afrag,
                                        int colBase, int n, int half) {
  const float* brow = X + (size_t)(colBase + n) * DIM + half * 2;
  v2f b[DIM / 4];
#pragma unroll
  for (int kk = 0; kk < DIM / 4; ++kk)
    b[kk] = *(const v2f*)(brow + kk * 4);

  v8f acc0 = {};
  v8f acc1 = {};
#pragma unroll
  for (int kk = 0; kk < DIM / 4; kk += 2) {
    acc0 = __builtin_amdgcn_wmma_f32_16x16x4_f32(false, afrag[kk], false, b[kk],
                                                 (short)0, acc0, false, false);
    acc1 = __builtin_amdgcn_wmma_f32_16x16x4_f32(false, afrag[kk + 1], false, b[kk + 1],
                                                 (short)0, acc1, false, false);
  }
#pragma unroll
  for (int r = 0; r < 8; ++r) acc0[r] += acc1[r];
  return acc0;
}

// One block (256 thr = 8 waves) owns a 16-row tile; sweeps all column tiles twice.
__global__ __launch_bounds__(256) void hm_main(const float* __restrict__ X,
                                               const int* __restrict__ T,
                                               float* __restrict__ negsum,
                                               float* __restrict__ negcnt,
                                               float* __restrict__ possum,
                                               float* __restrict__ poscnt,
                                               float* __restrict__ scal) {
  __shared__ float red[16 * 128];
  __shared__ float s_minpos[16];
  __shared__ float s_maxneg[16];

  const int tid  = threadIdx.x;
  const int wave = tid >> 5;
  const int lane = tid & 31;
  const int n    = lane & 15;
  const int half = lane >> 4;
  const int rowTile = blockIdx.x;
  const int rowBase = rowTile * 16;
  const int slot = wave * 16 + n;  // 128 partial slots per row

  // Preload A fragment (this block's 16 rows x 128 K, striped per WMMA layout).
  v2f afrag[DIM / 4];
  {
    const float* arow = X + (size_t)(rowBase + n) * DIM;
#pragma unroll
    for (int kk = 0; kk < DIM / 4; ++kk)
      afrag[kk] = *(const v2f*)(arow + kk * 4 + half * 2);
  }

  // Row targets for the 8 C-rows this thread produces (C row = r + 8*half).
  int ti[8];
#pragma unroll
  for (int r = 0; r < 8; ++r) ti[r] = T[rowBase + r + 8 * half];

  // ---------------- Pass A: per-row min over positives, max over negatives ---
  float mn[8], mx[8];
#pragma unroll
  for (int r = 0; r < 8; ++r) { mn[r] = 1e30f; mx[r] = -1e30f; }
  // Row N-1 full pos/neg sums (only nonzero in block NTILE-1, r==7, half==1).
  float pall = 0.f, pallc = 0.f, nall = 0.f, nallc = 0.f;

  for (int ct = wave; ct < NTILE; ct += 8) {
    const int colBase = ct * 16;
    // Warm L2/L0 for this wave's next tile (gfx1250 global_prefetch_b8).
    if (ct + 8 < NTILE)
      __builtin_prefetch(X + (size_t)((ct + 8) * 16 + n) * DIM, 0, 0);
    const int j  = colBase + n;
    const int tj = T[j];
    v8f acc = sim_tile(X, afrag, colBase, n, half);
#pragma unroll
    for (int r = 0; r < 8; ++r) {
      const int   i = rowBase + r + 8 * half;
      const float s = acc[r];
      const bool pos = (tj == ti[r]) && (i != j);
      const bool neg = (tj != ti[r]);
      if (pos) mn[r] = fminf(mn[r], s);
      if (neg) mx[r] = fmaxf(mx[r], s);
      if (i == NROW - 1) {
        if (pos) { pall += s; pallc += 1.f; }
        if (neg) { nall += s; nallc += 1.f; }
      }
    }
  }

  // Deterministic cross-wave reduction (one thread per row, fixed order).
#pragma unroll
  for (int r = 0; r < 8; ++r) red[(r + 8 * half) * 128 + slot] = mn[r];
  __syncthreads();
  if (tid < 16) {
    float v = 1e30f;
    for (int s2 = 0; s2 < 128; ++s2) v = fminf(v, red[tid * 128 + s2]);
    s_minpos[tid] = v;
  }
  __syncthreads();
#pragma unroll
  for (int r = 0; r < 8; ++r) red[(r + 8 * half) * 128 + slot] = mx[r];
  __syncthreads();
  if (tid < 16) {
    float v = -1e30f;
    for (int s2 = 0; s2 < 128; ++s2) v = fmaxf(v, red[tid * 128 + s2]);
    s_maxneg[tid] = v;
  }
  __syncthreads();

  // Block owning row N-1: reduce the 4 last-row scalars deterministically.
  if (rowTile == NTILE - 1) {
    float vals[4];
    vals[0] = pall; vals[1] = pallc; vals[2] = nall; vals[3] = nallc;
#pragma unroll
    for (int q = 0; q < 4; ++q) {
      __syncthreads();
      red[tid] = vals[q];
      __syncthreads();
      if (tid == 0) {
        float v = 0.f;
        for (int s2 = 0; s2 < 256; ++s2) v += red[s2];
        scal[q] = v;
      }
    }
    __syncthreads();
  }

  // ---------------- Pass B: thresholded sums/counts --------------------------
  float thrN[8], thrP[8];
#pragma unroll
  for (int r = 0; r < 8; ++r) {
    thrN[r] = s_minpos[r + 8 * half] - MARGIN;  // neg_sel: s >  thrN
    thrP[r] = s_maxneg[r + 8 * half] + MARGIN;  // pos_sel: s <  thrP
  }
  float ns[8], nc[8], ps[8], pc[8];
#pragma unroll
  for (int r = 0; r < 8; ++r) { ns[r] = 0.f; nc[r] = 0.f; ps[r] = 0.f; pc[r] = 0.f; }

  for (int ct = wave; ct < NTILE; ct += 8) {
    const int colBase = ct * 16;
    if (ct + 8 < NTILE)
      __builtin_prefetch(X + (size_t)((ct + 8) * 16 + n) * DIM, 0, 0);
    const int j  = colBase + n;
    const int tj = T[j];
    v8f acc = sim_tile(X, afrag, colBase, n, half);
#pragma unroll
    for (int r = 0; r < 8; ++r) {
      const int   i = rowBase + r + 8 * half;
      const float s = acc[r];
      const bool pos = (tj == ti[r]) && (i != j);
      const bool neg = (tj != ti[r]);
      if (neg && s > thrN[r]) { ns[r] += s;       nc[r] += 1.f; }
      if (pos && s < thrP[r]) { ps[r] += 1.f - s; pc[r] += 1.f; }
    }
  }

  // Deterministic reduction + writeout of the four per-row quantities.
  __syncthreads();
#pragma unroll
  for (int r = 0; r < 8; ++r) red[(r + 8 * half) * 128 + slot] = ns[r];
  __syncthreads();
  if (tid < 16) {
    float v = 0.f;
    for (int s2 = 0; s2 < 128; ++s2) v += red[tid * 128 + s2];
    negsum[rowBase + tid] = v;
  }
  __syncthreads();
#pragma unroll
  for (int r = 0; r < 8; ++r) red[(r + 8 * half) * 128 + slot] = nc[r];
  __syncthreads();
  if (tid < 16) {
    float v = 0.f;
    for (int s2 = 0; s2 < 128; ++s2) v += red[tid * 128 + s2];
    negcnt[rowBase + tid] = v;
  }
  __syncthreads();
#pragma unroll
  for (int r = 0; r < 8; ++r) red[(r + 8 * half) * 128 + slot] = ps[r];
  __syncthreads();
  if (tid < 16) {
    float v = 0.f;
    for (int s2 = 0; s2 < 128; ++s2) v += red[tid * 128 + s2];
    possum[rowBase + tid] = v;
  }
  __syncthreads();
#pragma unroll
  for (int r = 0; r < 8; ++r) red[(r + 8 * half) * 128 + slot] = pc[r];
  __syncthreads();
  if (tid < 16) {
    float v = 0.f;
    for (int s2 = 0; s2 < 128; ++s2) v += red[tid * 128 + s2];
    poscnt[rowBase + tid] = v;
  }
}

// Final fold: per-row losses -> (loss, prec, mean_pos_sim, mean_neg_sim).
__global__ __launch_bounds__(256) void hm_final(const float* __restrict__ negsum,
                                                const float* __restrict__ negcnt,
                                                const float* __restrict__ possum,
                                                const float* __restrict__ poscnt,
                                                const float* __restrict__ scal,
                                                float* __restrict__ out) {
  __shared__ float sl[256];
  __shared__ float sc[256];
  const int tid = threadIdx.x;
  float ls = 0.f, cs = 0.f;
  for (int r = tid * (NROW / 256); r < (tid + 1) * (NROW / 256); ++r) {
    const float ncv = negcnt[r];
    const float pcv = poscnt[r];
    const float nl = negsum[r] / fmaxf(ncv, 1.f);
    const float pl = possum[r] / fmaxf(pcv, 1.f);
    const bool valid = (ncv >= 1.f);
    ls += valid ? (pl + nl) : 0.f;
    cs += valid ? 0.f : 1.f;
  }
  sl[tid] = ls; sc[tid] = cs;
  __syncthreads();
  if (tid == 0) {
    float L = 0.f, C = 0.f;
    for (int s = 0; s < 256; ++s) { L += sl[s]; C += sc[s]; }
    out[0] = L / (float)NROW;                   // loss
    out[1] = C / (float)NROW;                   // prec = c/n
    out[2] = scal[0] / fmaxf(scal[1], 1.f);     // mean_pos_sim (row n-1)
    out[3] = scal[2] / fmaxf(scal[3], 1.f);     // mean_neg_sim (row n-1)
  }
}

extern "C" void kernel_launch(void* const* d_in, const int* in_sizes, int n_in,
                              void* d_out, int out_size, void* d_ws, size_t ws_size,
                              hipStream_t stream) {
  (void)in_sizes; (void)n_in; (void)out_size; (void)ws_size;
  const float* X = (const float*)d_in[0];
  const int*   T = (const int*)d_in[1];

  float* ws     = (float*)d_ws;
  float* negsum = ws;
  float* negcnt = ws + NROW;
  float* possum = ws + 2 * NROW;
  float* poscnt = ws + 3 * NROW;
  float* scal   = ws + 4 * NROW;   // 4 scalars: pall_sum, pall_cnt, nall_sum, nall_cnt

  hm_main<<<NTILE, 256, 0, stream>>>(X, T, negsum, negcnt, possum, poscnt, scal);
  hm_final<<<1, 256, 0, stream>>>(negsum, negcnt, possum, poscnt, scal, (float*)d_out);
}